// GCN2Conv_936302871062
// MI455X (gfx1250) — compile-verified
//
#include <hip/hip_runtime.h>
#include <hip/hip_bf16.h>
#include <math.h>

#define CH 128
#define SS_STRIDE 132             // LDS dword stride for S tile (even -> b64-aligned; 4k+j banking)
#define WT2_STRIDE 130            // float2 stride for paired W^T rows
#define WT2_SKEW 14               // odd-kk row skew (float2) -> half-wave reads land 32 banks apart
#define ALPHA_C 0.1f
#define BETA_C 0.22314355131420976f   // log(1/4 + 1)
#define ONE_MINUS_BETA_C 0.7768564486857903f

typedef float v2f __attribute__((ext_vector_type(2)));
typedef float v8f __attribute__((ext_vector_type(8)));

__device__ __forceinline__ int wt2_idx(int kk, int col) {   // float2 units
    return kk * WT2_STRIDE + (kk & 1) * WT2_SKEW + col;
}

// ---------------- degree / norm kernels ----------------

__global__ void k_init_deg(float* __restrict__ deg, int N) {
    int i = blockIdx.x * blockDim.x + threadIdx.x;
    if (i < N) deg[i] = 1.0f;                       // self-loop
}

__global__ void k_count_deg(const int* __restrict__ dst, float* __restrict__ deg, int E) {
    int e = blockIdx.x * blockDim.x + threadIdx.x;
    if (e < E) atomicAdd(&deg[dst[e]], 1.0f);       // exact: integer-valued f32 adds
}

__global__ void k_dinv(float* __restrict__ deg, int N) {
    int i = blockIdx.x * blockDim.x + threadIdx.x;
    if (i < N) deg[i] = rsqrtf(deg[i]);             // deg >= 1 always
}

// propagated init with self-loop term: out = x * dinv^2   (float4 streaming)
__global__ void k_selfloop(const float* __restrict__ x, const float* __restrict__ dinv,
                           float* __restrict__ out, int N) {
    int i = blockIdx.x * blockDim.x + threadIdx.x;  // one float4 per thread
    int total = N * (CH / 4);
    if (i >= total) return;
    int row = i >> 5;                               // CH/4 == 32 float4 per row
    float w = dinv[row];
    w = w * w;
    float4 v = reinterpret_cast<const float4*>(x)[i];
    float4 o = { v.x * w, v.y * w, v.z * w, v.w * w };
    reinterpret_cast<float4*>(out)[i] = o;
}

// scatter-add of normalized messages: 1 edge per 32 lanes, 4 atomics/lane
__global__ __launch_bounds__(256) void k_edge_scatter(
    const int* __restrict__ src, const int* __restrict__ dst,
    const float* __restrict__ x, const float* __restrict__ dinv,
    float* __restrict__ acc, int E)
{
    int e = blockIdx.x * 8 + (threadIdx.x >> 5);
    if (e >= E) return;
    int lane = threadIdx.x & 31;
    int s = src[e];
    int d = dst[e];
    float norm = dinv[s] * dinv[d];
    const float* xs = x + (size_t)s * CH;
    float* ad = acc + (size_t)d * CH;
#pragma unroll
    for (int q = 0; q < 4; ++q) {
        int c = lane + q * 32;
        atomicAdd(ad + c, xs[c] * norm);            // global_atomic_add_f32, resolves in L2
    }
}

// ---------------- fused support + GEMM(W^T) + epilogue, f32 WMMA ----------------
// Block: 256 threads = 8 waves, 128 rows x 128 cols tile.
// Wave w computes rows [w*16, w*16+16) x all 128 cols as 8 jt-tiles of 16x16,
// accumulating over K=128 in 32 steps of v_wmma_f32_16x16x4_f32.
__global__ __launch_bounds__(256) void k_gemm_epilogue(
    const float* __restrict__ x0, const float* __restrict__ W,
    const float* __restrict__ bias, float* __restrict__ out, int N)
{
    __shared__ float sS[CH * SS_STRIDE];                      // support tile S[r][k]
    __shared__ float sWt2[(64 * WT2_STRIDE + WT2_SKEW) * 2];  // paired {Wt[2kk], Wt[2kk+1]}[col]

    const int tid  = threadIdx.x;
    const int wave = tid >> 5;
    const int lane = tid & 31;
    const int half = lane >> 4;                     // 0: lanes 0-15, 1: lanes 16-31
    const int l16  = lane & 15;
    const int row0 = blockIdx.x * CH;

    // Stage W transposed+K-paired into LDS (coalesced global read; 2-way LDS write conflict, one-shot).
    for (int i = tid; i < CH * CH; i += 256) {
        int j = i >> 7;                             // W row  (output channel -> B column)
        int k = i & (CH - 1);                       // W col  (input channel  -> B row / K)
        sWt2[2 * wt2_idx(k >> 1, j) + (k & 1)] = W[i];
    }
    // Stage S = (1-alpha)*propagated + alpha*x0 (propagated currently lives in `out`).
    for (int i = tid; i < CH * (CH / 4); i += 256) {
        int r  = i >> 5;                            // row within tile
        int c4 = i & 31;                            // float4 column
        int row = row0 + r;
        int rr  = row < N ? row : (N - 1);          // clamp tail (uniform trip count)
        float4 p = reinterpret_cast<const float4*>(out)[(size_t)rr * 32 + c4];
        float4 z = reinterpret_cast<const float4*>(x0)[(size_t)rr * 32 + c4];
        float4 s = { (1.0f - ALPHA_C) * p.x + ALPHA_C * z.x,
                     (1.0f - ALPHA_C) * p.y + ALPHA_C * z.y,
                     (1.0f - ALPHA_C) * p.z + ALPHA_C * z.z,
                     (1.0f - ALPHA_C) * p.w + ALPHA_C * z.w };
        reinterpret_cast<float4*>(&sS[r * SS_STRIDE])[c4] = s;
    }
    __syncthreads();

    const int arow = wave * 16 + l16;               // A-fragment row (ISA 16x4 f32 layout)

    v8f acc[8] = {};                                // 8 column tiles, 8 VGPRs each
#pragma unroll 4
    for (int k0 = 0; k0 < CH; k0 += 4) {
        const int ka = k0 + 2 * half;               // lanes 0-15: K=k0,k0+1 ; 16-31: K=k0+2,k0+3
        const int kk = (k0 >> 1) + half;            // paired-row index for B
        v2f a = *reinterpret_cast<const v2f*>(&sS[arow * SS_STRIDE + ka]);   // ds_load_b64
        (void)ka;
#pragma unroll
        for (int jt = 0; jt < 8; ++jt) {
            const int col = jt * 16 + l16;
            v2f bfrag = *reinterpret_cast<const v2f*>(&sWt2[2 * wt2_idx(kk, col)]); // ds_load_b64
            acc[jt] = __builtin_amdgcn_wmma_f32_16x16x4_f32(
                /*neg_a=*/false, a, /*neg_b=*/false, bfrag,
                /*c_mod=*/(short)0, acc[jt], /*reuse_a=*/false, /*reuse_b=*/false);
        }
    }

    // Epilogue: out = (1-beta)*S + beta*(D + b). D layout: VGPR i -> M=i (lanes0-15)/M=8+i.
#pragma unroll
    for (int jt = 0; jt < 8; ++jt) {
        const int col = jt * 16 + l16;
        const float bb = bias[col];
#pragma unroll
        for (int i = 0; i < 8; ++i) {
            const int m   = i + 8 * half;
            const int row = row0 + wave * 16 + m;
            if (row < N) {
                float s = sS[(wave * 16 + m) * SS_STRIDE + col];
                out[(size_t)row * CH + col] =
                    ONE_MINUS_BETA_C * s + BETA_C * (acc[jt][i] + bb);
            }
        }
    }
}

// ---------------- launcher ----------------

extern "C" void kernel_launch(void* const* d_in, const int* in_sizes, int n_in,
                              void* d_out, int out_size, void* d_ws, size_t ws_size,
                              hipStream_t stream) {
    const float* x    = (const float*)d_in[0];
    const float* x0   = (const float*)d_in[1];
    const float* W    = (const float*)d_in[2];
    const float* bias = (const float*)d_in[3];
    const int*   edge = (const int*)d_in[4];

    const int N = in_sizes[0] / CH;
    const int E = in_sizes[4] / 2;
    const int* src = edge;          // edge_index[0]
    const int* dst = edge + E;      // edge_index[1]

    float* out  = (float*)d_out;    // doubles as the propagate accumulator
    float* dinv = (float*)d_ws;     // N floats of scratch (deg -> dinv in place)

    k_init_deg <<<(N + 255) / 256, 256, 0, stream>>>(dinv, N);
    k_count_deg<<<(E + 255) / 256, 256, 0, stream>>>(dst, dinv, E);
    k_dinv     <<<(N + 255) / 256, 256, 0, stream>>>(dinv, N);

    const int q4 = N * (CH / 4);
    k_selfloop<<<(q4 + 255) / 256, 256, 0, stream>>>(x, dinv, out, N);
    k_edge_scatter<<<(E + 7) / 8, 256, 0, stream>>>(src, dst, x, dinv, out, E);

    k_gemm_epilogue<<<(N + CH - 1) / CH, 256, 0, stream>>>(x0, W, bias, out, N);
}